// _FallbackSurrogateGNN_73873437491768
// MI455X (gfx1250) — compile-verified
//
#include <hip/hip_runtime.h>

// ---- Problem constants (from reference) ----
#define N_NODES  100000
#define N_EDGES  1250000
#define G_GRAPHS 64
#define F_IN_DIM 29
#define H_DIM    64
#define NT_DIM   6
#define GT_DIM   4
#define L_LAYERS 4

typedef float v2f __attribute__((ext_vector_type(2)));
typedef float v8f __attribute__((ext_vector_type(8)));

// Non-returning hardware float atomic (global_atomic_add_f32, STOREcnt path).
__device__ __forceinline__ void atomic_add_f32(float* p, float v) {
  (void)__hip_atomic_fetch_add(p, v, __ATOMIC_RELAXED, __HIP_MEMORY_SCOPE_AGENT);
}

// ---------------------------------------------------------------------------
// deg starts at 1.0 (self loop); gsum/gcnt zeroed for graph pooling.
__global__ void k_init(float* __restrict__ deg, float* __restrict__ gsum,
                       float* __restrict__ gcnt) {
  int i = blockIdx.x * blockDim.x + threadIdx.x;
  if (i < N_NODES) deg[i] = 1.0f;
  if (i < G_GRAPHS * H_DIM) gsum[i] = 0.0f;
  if (i < G_GRAPHS) gcnt[i] = 0.0f;
}

// deg[col[e]] += 1 for every real edge.
__global__ void k_degree(const int* __restrict__ col, float* __restrict__ deg) {
  int e = blockIdx.x * blockDim.x + threadIdx.x;
  if (e < N_EDGES) atomic_add_f32(&deg[col[e]], 1.0f);
}

// dis = deg^-0.5 in place (deg >= 1 always due to self loops).
__global__ void k_rsqrt(float* __restrict__ dis) {
  int i = blockIdx.x * blockDim.x + threadIdx.x;
  if (i < N_NODES) dis[i] = rsqrtf(dis[i]);
}

// ---------------------------------------------------------------------------
// Encoder: h = relu(x @ encW + encb).  K=29 is awkward for WMMA and the op is
// ~0.4 GFLOP total, so plain VALU with encW staged in LDS.
__global__ void __launch_bounds__(256)
k_encoder(const float* __restrict__ x, const float* __restrict__ encW,
          const float* __restrict__ encb, float* __restrict__ h) {
  __shared__ float sW[F_IN_DIM * H_DIM];
  __shared__ float sb[H_DIM];
  for (int i = threadIdx.x; i < F_IN_DIM * H_DIM; i += 256) sW[i] = encW[i];
  if (threadIdx.x < H_DIM) sb[threadIdx.x] = encb[threadIdx.x];
  __syncthreads();
  int idx = blockIdx.x * blockDim.x + threadIdx.x;
  if (idx >= N_NODES * H_DIM) return;
  int n = idx >> 6, f = idx & 63;
  const float* xr = x + (size_t)n * F_IN_DIM;
  float acc = sb[f];
#pragma unroll
  for (int k = 0; k < F_IN_DIM; ++k) acc = fmaf(xr[k], sW[k * H_DIM + f], acc);
  h[idx] = fmaxf(acc, 0.0f);
}

// ---------------------------------------------------------------------------
// hw = h @ W  (N x 64 @ 64 x 64), full f32 via V_WMMA_F32_16X16X4_F32.
// Block = 256 threads = 8 waves; wave (wq, ct) owns output tile
// rows [ (2*blockIdx.x+wq)*16 , +16 ), cols [ ct*16, +16 ).  K=64 -> 16 wmma.
// ISA layouts (wave32):
//   A 16x4 : lane L, vgpr v -> A[L%16][k0 + 2*(L/16) + v]   (contig float2)
//   B 4x16 : lane L, vgpr v -> B[k0 + 2*(L/16) + v][L%16]
//   D 16x16: lane L, vgpr v -> D[v + 8*(L/16)][L%16]
__global__ void __launch_bounds__(256)
k_gemm64(const float* __restrict__ hin, const float* __restrict__ W,
         float* __restrict__ hw) {
  __shared__ float sW[H_DIM * H_DIM];
  for (int i = threadIdx.x; i < H_DIM * H_DIM; i += 256) sW[i] = W[i];
  __syncthreads();

  const int wave = threadIdx.x >> 5;
  const int lane = threadIdx.x & 31;
  const int half = lane >> 4;     // 0: lanes 0-15, 1: lanes 16-31
  const int mrow = lane & 15;

  const int row0 = (blockIdx.x * 2 + (wave >> 2)) * 16;  // node tile
  const int col0 = (wave & 3) * 16;                      // output-col tile

  v8f c = {};
  const float* aptr = hin + (size_t)(row0 + mrow) * H_DIM + 2 * half;
#pragma unroll
  for (int kk = 0; kk < 16; ++kk) {
    const int k0 = kk * 4;
    v2f a = *(const v2f*)(aptr + k0);
    v2f b;
    b.x = sW[(k0 + 2 * half) * H_DIM + col0 + mrow];
    b.y = sW[(k0 + 2 * half + 1) * H_DIM + col0 + mrow];
    c = __builtin_amdgcn_wmma_f32_16x16x4_f32(false, a, false, b,
                                              (short)0, c, false, false);
  }
  float* dptr = hw + (size_t)(row0 + 8 * half) * H_DIM + col0 + mrow;
#pragma unroll
  for (int v = 0; v < 8; ++v) dptr[(size_t)v * H_DIM] = c[v];
}

// ---------------------------------------------------------------------------
// agg = hw * dis^2   (self-loop contribution; also initializes agg).
// float4-vectorized: idx covers N*16 float4s.
__global__ void k_selfloop(const float4* __restrict__ hw,
                           const float* __restrict__ dis,
                           float4* __restrict__ agg) {
  int idx = blockIdx.x * blockDim.x + threadIdx.x;
  if (idx >= N_NODES * (H_DIM / 4)) return;
  float d = dis[idx >> 4];
  float d2 = d * d;
  float4 v = hw[idx];
  v.x *= d2; v.y *= d2; v.z *= d2; v.w *= d2;
  agg[idx] = v;
}

// agg[col] += hw[row] * dis[row]*dis[col].  64 consecutive threads share one
// edge -> coalesced gather of hw row and cacheline-grouped atomics on agg row.
__global__ void __launch_bounds__(256)
k_scatter(const int* __restrict__ row, const int* __restrict__ col,
          const float* __restrict__ dis, const float* __restrict__ hw,
          float* __restrict__ agg) {
  int idx = blockIdx.x * blockDim.x + threadIdx.x;  // < E*64 = 80M < 2^31
  if (idx >= N_EDGES * H_DIM) return;
  int e = idx >> 6, f = idx & 63;
  int r = row[e], c = col[e];
  float nrm = dis[r] * dis[c];
  atomic_add_f32(&agg[(size_t)c * H_DIM + f], hw[(size_t)r * H_DIM + f] * nrm);
}

// h = relu(agg + convb)   (float4-vectorized)
__global__ void k_relubias(const float4* __restrict__ agg,
                           const float* __restrict__ b, float4* __restrict__ h) {
  int idx = blockIdx.x * blockDim.x + threadIdx.x;
  if (idx >= N_NODES * (H_DIM / 4)) return;
  int f4 = (idx & 15) * 4;
  float4 v = agg[idx];
  v.x = fmaxf(v.x + b[f4 + 0], 0.0f);
  v.y = fmaxf(v.y + b[f4 + 1], 0.0f);
  v.z = fmaxf(v.z + b[f4 + 2], 0.0f);
  v.w = fmaxf(v.w + b[f4 + 3], 0.0f);
  h[idx] = v;
}

// ---------------------------------------------------------------------------
// pred_node = h @ nodeW + nodeb  -> d_out[0 .. N*NT)
__global__ void k_prednode(const float* __restrict__ h,
                           const float* __restrict__ nodeW,
                           const float* __restrict__ nodeb,
                           float* __restrict__ out) {
  int idx = blockIdx.x * blockDim.x + threadIdx.x;
  if (idx >= N_NODES * NT_DIM) return;
  int n = idx / NT_DIM, t = idx % NT_DIM;
  const float* hr = h + (size_t)n * H_DIM;
  float acc = nodeb[t];
#pragma unroll 8
  for (int k = 0; k < H_DIM; ++k) acc = fmaf(hr[k], nodeW[k * NT_DIM + t], acc);
  out[idx] = acc;
}

// Graph pooling accumulation: gsum[batch[n]] += h[n]; gcnt[batch[n]] += 1.
__global__ void k_pool(const float* __restrict__ h, const int* __restrict__ batch,
                       float* __restrict__ gsum, float* __restrict__ gcnt) {
  int idx = blockIdx.x * blockDim.x + threadIdx.x;
  if (idx >= N_NODES * H_DIM) return;
  int n = idx >> 6, f = idx & 63;
  int b = batch[n];
  atomic_add_f32(&gsum[b * H_DIM + f], h[idx]);
  if (f == 0) atomic_add_f32(&gcnt[b], 1.0f);
}

// pred_global = relu(gemb@g1W+g1b) @ g2W + g2b  (single block; tiny)
__global__ void __launch_bounds__(256)
k_ghead(const float* __restrict__ gsum, const float* __restrict__ gcnt,
        const float* __restrict__ g1W, const float* __restrict__ g1b,
        const float* __restrict__ g2W, const float* __restrict__ g2b,
        float* __restrict__ out) {
  __shared__ float emb[G_GRAPHS * H_DIM];
  __shared__ float t1[G_GRAPHS * H_DIM];
  for (int i = threadIdx.x; i < G_GRAPHS * H_DIM; i += 256) {
    emb[i] = gsum[i] / fmaxf(gcnt[i >> 6], 1.0f);
  }
  __syncthreads();
  for (int i = threadIdx.x; i < G_GRAPHS * H_DIM; i += 256) {
    int g = i >> 6, j = i & 63;
    float acc = g1b[j];
    for (int k = 0; k < H_DIM; ++k)
      acc = fmaf(emb[g * H_DIM + k], g1W[k * H_DIM + j], acc);
    t1[i] = fmaxf(acc, 0.0f);
  }
  __syncthreads();
  // 256 threads == 64 graphs * 4 outputs
  int g = threadIdx.x >> 2, t = threadIdx.x & 3;
  float acc = g2b[t];
  for (int k = 0; k < H_DIM; ++k)
    acc = fmaf(t1[g * H_DIM + k], g2W[k * GT_DIM + t], acc);
  out[(size_t)N_NODES * NT_DIM + threadIdx.x] = acc;
}

// ---------------------------------------------------------------------------
extern "C" void kernel_launch(void* const* d_in, const int* in_sizes, int n_in,
                              void* d_out, int out_size, void* d_ws, size_t ws_size,
                              hipStream_t stream) {
  const float* x     = (const float*)d_in[0];
  const int*   ei    = (const int*)d_in[1];   // (2, E) int32
  const int*   batch = (const int*)d_in[2];
  const float* encW  = (const float*)d_in[3];
  const float* encb  = (const float*)d_in[4];
  const float* convW = (const float*)d_in[5]; // (L, 64, 64)
  const float* convb = (const float*)d_in[6]; // (L, 64)
  const float* nodeW = (const float*)d_in[7];
  const float* nodeb = (const float*)d_in[8];
  const float* g1W   = (const float*)d_in[9];
  const float* g1b   = (const float*)d_in[10];
  const float* g2W   = (const float*)d_in[11];
  const float* g2b   = (const float*)d_in[12];
  float* out = (float*)d_out;

  // Workspace layout (floats): dis | h | hw | agg | gsum | gcnt
  float* ws   = (float*)d_ws;
  float* dis  = ws;
  float* h    = dis + N_NODES;
  float* hw   = h   + (size_t)N_NODES * H_DIM;
  float* agg  = hw  + (size_t)N_NODES * H_DIM;
  float* gsum = agg + (size_t)N_NODES * H_DIM;
  float* gcnt = gsum + G_GRAPHS * H_DIM;

  const int* row = ei;            // edge_index[0] (senders)
  const int* col = ei + N_EDGES;  // edge_index[1] (receivers)

  const int B = 256;
  const int gN    = (N_NODES + B - 1) / B;                    // 391
  const int gE    = (N_EDGES + B - 1) / B;                    // 4883
  const int gNH   = (N_NODES * H_DIM + B - 1) / B;            // 25000
  const int gNH4  = (N_NODES * (H_DIM / 4) + B - 1) / B;      // 6250
  const int gEH   = (N_EDGES * H_DIM + B - 1) / B;            // 312500
  const int gGEMM = N_NODES / (2 * 16);                       // 3125 (exact)
  const int gPN   = (N_NODES * NT_DIM + B - 1) / B;           // 2344

  k_init<<<gN, B, 0, stream>>>(dis, gsum, gcnt);
  k_degree<<<gE, B, 0, stream>>>(col, dis);
  k_rsqrt<<<gN, B, 0, stream>>>(dis);
  k_encoder<<<gNH, B, 0, stream>>>(x, encW, encb, h);

  for (int l = 0; l < L_LAYERS; ++l) {
    k_gemm64<<<gGEMM, B, 0, stream>>>(h, convW + (size_t)l * H_DIM * H_DIM, hw);
    k_selfloop<<<gNH4, B, 0, stream>>>((const float4*)hw, dis, (float4*)agg);
    k_scatter<<<gEH, B, 0, stream>>>(row, col, dis, hw, agg);
    k_relubias<<<gNH4, B, 0, stream>>>((const float4*)agg,
                                       convb + (size_t)l * H_DIM, (float4*)h);
  }

  k_prednode<<<gPN, B, 0, stream>>>(h, nodeW, nodeb, out);
  k_pool<<<gNH, B, 0, stream>>>(h, batch, gsum, gcnt);
  k_ghead<<<1, B, 0, stream>>>(gsum, gcnt, g1W, g1b, g2W, g2b, out);
}